// AttributedEncoder_26620207300627
// MI455X (gfx1250) — compile-verified
//
#include <hip/hip_runtime.h>
#include <math.h>

#define N_ENT 50000
#define N_ATT 2000
#define N_VAL 100000
#define NE    300000
#define KD    256   // KEY_DIM == VAL_DIM

typedef __attribute__((ext_vector_type(2))) float v2f;
typedef __attribute__((ext_vector_type(8))) float v8f;

// ---------------------------------------------------------------- zero fill
__global__ void zero_f32(float* __restrict__ p, long n) {
    long i = (long)blockIdx.x * blockDim.x + threadIdx.x;
    long stride = (long)gridDim.x * blockDim.x;
    for (; i < n; i += stride) p[i] = 0.0f;
}

// ------------------------------------------------- per-row dot with a_w halves
// rows [0, N_ENT): ent_score[r] = dot(ent_feats[r], a_w[0:256])
// rows [N_ENT, N_ENT+N_ATT): att_score[r-N_ENT] = dot(att_feats[r-N_ENT], a_w[256:512])
__global__ void row_dot_kernel(const float* __restrict__ ent_feats,
                               const float* __restrict__ att_feats,
                               const float* __restrict__ a_w,
                               float* __restrict__ ent_score,
                               float* __restrict__ att_score) {
    int wave = (int)((blockIdx.x * blockDim.x + threadIdx.x) >> 5);
    int lane = threadIdx.x & 31;
    const float* feat;
    const float* w;
    float* outp;
    if (wave < N_ENT) {
        feat = ent_feats + (long)wave * KD;
        w = a_w;
        outp = ent_score + wave;
    } else {
        int r = wave - N_ENT;
        if (r >= N_ATT) return;
        feat = att_feats + (long)r * KD;
        w = a_w + KD;
        outp = att_score + r;
    }
    float s = 0.0f;
#pragma unroll
    for (int j = 0; j < 2; ++j) {
        int c = (lane + 32 * j) * 4;          // 32 lanes * 2 * float4 = 256
        float4 f = *(const float4*)(feat + c);
        float4 ww = *(const float4*)(w + c);
        s += f.x * ww.x + f.y * ww.y + f.z * ww.z + f.w * ww.w;
    }
#pragma unroll
    for (int off = 16; off > 0; off >>= 1) s += __shfl_xor(s, off, 32);
    if (lane == 0) *outp = s;
}

// ------------------------------------------------- per-edge score + row sums
__global__ void edge_score_kernel(const int* __restrict__ trip,
                                  const float* __restrict__ ent_score,
                                  const float* __restrict__ att_score,
                                  const float* __restrict__ a_b,
                                  float* __restrict__ score,
                                  float* __restrict__ row_sum) {
    int e = blockIdx.x * blockDim.x + threadIdx.x;
    if (e >= NE) return;
    int h  = trip[3 * e + 0];
    int at = trip[3 * e + 2];
    float x = ent_score[h] + att_score[at] + a_b[0];
    x = (x > 0.0f) ? x : 0.2f * x;            // leaky_relu(0.2)
    float sc = expf(x);
    score[e] = sc;
    atomicAdd(&row_sum[h], sc);
}

// ------------------------------------------------- WMMA fp32 GEMM, 16x16 tile/wave
// C[M x 256] (+)= A[M x 256] * B[256 x 256], all row-major fp32.
// One wave per 16x16 output tile; K consumed 4 at a time via V_WMMA_F32_16X16X4_F32.
// M must be a multiple of 16 and grid must supply exactly (M/16)*16 waves
// (EXEC all-ones required by WMMA).
template <bool ACC>
__global__ void wmma_gemm_f32(const float* __restrict__ A,
                              const float* __restrict__ B,
                              float* __restrict__ C) {
    int wave = (int)((blockIdx.x * blockDim.x + threadIdx.x) >> 5);
    int lane = threadIdx.x & 31;
    int tM = wave >> 4;                 // 16 tiles along N (256/16)
    int tN = wave & 15;
    int row0 = tM * 16;
    int col0 = tN * 16;

    // A 16x4 layout: lanes 0-15 hold K=k0+0,k0+1 ; lanes 16-31 hold K=k0+2,k0+3
    int m    = lane & 15;
    int koff = (lane >> 4) << 1;
    // B 4x16 layout: row K striped across lanes; same lane-half split in K
    int bn   = col0 + (lane & 15);
    // C/D layout: VGPR r -> row r (lanes 0-15) / row r+8 (lanes 16-31)
    int crow = row0 + ((lane >> 4) << 3);
    int cn   = col0 + (lane & 15);

    v8f c;
    if (ACC) {
#pragma unroll
        for (int r = 0; r < 8; ++r) c[r] = C[(long)(crow + r) * KD + cn];
    } else {
#pragma unroll
        for (int r = 0; r < 8; ++r) c[r] = 0.0f;
    }

    const float* aptr = A + (long)(row0 + m) * KD + koff;

#pragma unroll 4
    for (int k0 = 0; k0 < KD; k0 += 4) {
        v2f av = *(const v2f*)(aptr + k0);               // A[m, k0+koff .. +1]
        v2f bv;
        bv.x = B[(long)(k0 + koff + 0) * KD + bn];       // B[k0+koff,   bn]
        bv.y = B[(long)(k0 + koff + 1) * KD + bn];       // B[k0+koff+1, bn]
        c = __builtin_amdgcn_wmma_f32_16x16x4_f32(
                /*neg_a=*/false, av, /*neg_b=*/false, bv,
                /*c_mod=*/(short)0, c, /*reuse_a=*/false, /*reuse_b=*/false);
    }

#pragma unroll
    for (int r = 0; r < 8; ++r) C[(long)(crow + r) * KD + cn] = c[r];
}

// ------------------------------------------------- edge aggregation
// p_e = score[e]/row_sum[h];  U[h,:] += p_e * val_feats[val,:]
//                             out[h,:] += p_e * att_proj[att,:]
__global__ void edge_agg_kernel(const int* __restrict__ trip,
                                const float* __restrict__ score,
                                const float* __restrict__ row_sum,
                                const float* __restrict__ att_proj,
                                const float* __restrict__ val_feats,
                                float* __restrict__ U,
                                float* __restrict__ outAcc) {
    int d = threadIdx.x;                     // 256 threads = one feature dim each
    for (int e = blockIdx.x; e < NE; e += gridDim.x) {
        int h  = trip[3 * e + 0];
        int vl = trip[3 * e + 1];
        int at = trip[3 * e + 2];
        float p = score[e] / row_sum[h];
        atomicAdd(&U[(long)h * KD + d],      p * val_feats[(long)vl * KD + d]);
        atomicAdd(&outAcc[(long)h * KD + d], p * att_proj[(long)at * KD + d]);
    }
}

// ------------------------------------------------- finalize: elu(out + ent)
__global__ void finalize_kernel(float* __restrict__ out,
                                const float* __restrict__ ent_feats,
                                long n) {
    long i = (long)blockIdx.x * blockDim.x + threadIdx.x;
    long stride = (long)gridDim.x * blockDim.x;
    for (; i < n; i += stride) {
        float x = out[i] + ent_feats[i];
        out[i] = (x > 0.0f) ? x : (expf(x) - 1.0f);
    }
}

extern "C" void kernel_launch(void* const* d_in, const int* in_sizes, int n_in,
                              void* d_out, int out_size, void* d_ws, size_t ws_size,
                              hipStream_t stream) {
    const int*   trip      = (const int*)d_in[0];    // (E,3) = [h, val, att]
    const float* att_feats = (const float*)d_in[1];  // (N_ATT, 256)
    const float* val_feats = (const float*)d_in[2];  // (N_VAL, 256)
    const float* ent_feats = (const float*)d_in[3];  // (N_ENT, 256)
    const float* a_w       = (const float*)d_in[4];  // (1, 512)
    const float* a_b       = (const float*)d_in[5];  // (1,)
    const float* W         = (const float*)d_in[6];  // (512, 256)
    float* out = (float*)d_out;                      // (N_ENT, 256)

    const float* W_top = W;              // rows 0..255  -> applied to att features
    const float* W_bot = W + KD * KD;    // rows 256..511 -> applied to val features

    // Workspace layout (floats): ~55 MB total, fits in L2.
    float* ws       = (float*)d_ws;
    float* ent_s    = ws;                          // N_ENT
    float* att_s    = ent_s   + N_ENT;             // N_ATT
    float* row_sum  = att_s   + N_ATT;             // N_ENT
    float* score    = row_sum + N_ENT;             // NE
    float* att_proj = score   + NE;                // N_ATT * KD
    float* U        = att_proj + (long)N_ATT * KD; // N_ENT * KD

    const long nEntFeat = (long)N_ENT * KD;

    // 1) zero accumulators (row_sum, U, out)
    zero_f32<<<512, 256, 0, stream>>>(row_sum, (long)N_ENT);
    zero_f32<<<2048, 256, 0, stream>>>(U, nEntFeat);
    zero_f32<<<2048, 256, 0, stream>>>(out, nEntFeat);

    // 2) per-entity / per-attribute attention dot products (one wave per row)
    {
        int waves = N_ENT + N_ATT;                 // 52000 waves
        int blocks = (waves * 32 + 255) / 256;     // 6500 blocks of 8 waves
        row_dot_kernel<<<blocks, 256, 0, stream>>>(ent_feats, att_feats, a_w,
                                                   ent_s, att_s);
    }

    // 3) att_proj = att_feats @ W_top   (2000x256x256, WMMA fp32)
    {
        int tiles = (N_ATT / 16) * 16;             // 2000 wave-tiles
        wmma_gemm_f32<false><<<tiles / 8, 256, 0, stream>>>(att_feats, W_top, att_proj);
    }

    // 4) edge scores + softmax denominators
    edge_score_kernel<<<(NE + 255) / 256, 256, 0, stream>>>(trip, ent_s, att_s, a_b,
                                                            score, row_sum);

    // 5) aggregate: U[h] += p*val_feats[val];  out[h] += p*att_proj[att]
    edge_agg_kernel<<<2048, 256, 0, stream>>>(trip, score, row_sum,
                                              att_proj, val_feats, U, out);

    // 6) out += U @ W_bot   (50000x256x256, WMMA fp32, accumulate into out)
    {
        int tiles = (N_ENT / 16) * 16;             // 50000 wave-tiles
        wmma_gemm_f32<true><<<tiles / 8, 256, 0, stream>>>(U, W_bot, out);
    }

    // 7) out = elu(out + ent_feats)
    finalize_kernel<<<4096, 256, 0, stream>>>(out, ent_feats, nEntFeat);
}